// CapsulLayer_20916490731586
// MI455X (gfx1250) — compile-verified
//
#include <hip/hip_runtime.h>
#include <hip/hip_bf16.h>
#include <stdint.h>

typedef float v2f __attribute__((ext_vector_type(2)));
typedef float v8f __attribute__((ext_vector_type(8)));

#define NCAP   16
#define DCAP   32
#define NIN    1024
#define DIN    256
#define NOUT   512          // NCAP * DCAP
#define BATCH  128

// ---------------- GEMM: u_hat[b][cap][j][k] = sum_d x[b][j][d] * W[d][cap*32+k]
// Block: 256 threads (8 wave32), tile 256(M) x 64(N).
// Wave w computes rows [m0+32w, m0+32w+32) x cols [n0, n0+64): 2 A frags x 4 B frags.
// W tile staged in LDS in K-pair-interleaved layout so every B fragment is one
// aligned ds_load_b64:  wl[kp*130 + 2*col + (k&1)] = W[2*kp + (k&1)][n0+col]
#define BM 256
#define BN 64
#define KPS 130                 // dwords per K-pair row: 2*BN + 2 pad (bank de-alias)

__global__ __launch_bounds__(256)
void capsule_gemm(const float* __restrict__ x, const float* __restrict__ W,
                  float* __restrict__ uhat)
{
    __shared__ float wl[(DIN / 2) * KPS];   // 128 * 130 * 4 = 66,560 B
    const int n0  = blockIdx.x * BN;
    const int m0  = blockIdx.y * BM;
    const int b   = blockIdx.z;
    const int tid = threadIdx.x;

    // ---- async-stage W tile into interleaved LDS layout (DMA path, ASYNCcnt)
    for (int i = tid; i < DIN * BN; i += 256) {
        const int k = i >> 6, c = i & 63;
        const unsigned dst = (unsigned)(uintptr_t)&wl[(k >> 1) * KPS + 2 * c + (k & 1)];
        const int goff = (int)(((unsigned)(k * NOUT + n0 + c)) * 4u);
        asm volatile("global_load_async_to_lds_b32 %0, %1, %2"
                     :: "v"(dst), "v"(goff), "s"(W) : "memory");
    }
    asm volatile("s_wait_asynccnt 0x0" ::: "memory");
    __syncthreads();

    const int wave = tid >> 5;
    const int lane = tid & 31;
    const int half = lane >> 4;             // lane group g: K-pair selector
    const int l16  = lane & 15;

    // A rows (A 16x4 fp32 layout: lanes 0-15 M=0..15 K={0,1}; lanes 16-31 K={2,3})
    const float* xr0 = x + ((size_t)b * NIN + (size_t)(m0 + wave * 32 + l16)) * DIN;
    const float* xr1 = xr0 + 16 * DIN;
    __builtin_prefetch(xr0, 0, 3);
    __builtin_prefetch(xr1, 0, 3);

    v8f acc[2][4] = {};

    #pragma unroll 2
    for (int k0 = 0; k0 < DIN; k0 += 4) {
        const v2f a0 = *(const v2f*)(xr0 + k0 + 2 * half);   // 8B global b64
        const v2f a1 = *(const v2f*)(xr1 + k0 + 2 * half);

        // B frag (rows k0+2g, k0+2g+1 at col l16 + 16t) = one ds_load_b64 each
        const float* wk = wl + ((k0 >> 1) + half) * KPS + 2 * l16;
        v2f bf[4];
        #pragma unroll
        for (int t = 0; t < 4; ++t) bf[t] = *(const v2f*)(wk + 32 * t);

        #pragma unroll
        for (int t = 0; t < 4; ++t) {
            acc[0][t] = __builtin_amdgcn_wmma_f32_16x16x4_f32(false, a0, false, bf[t],
                                                              (short)0, acc[0][t], false, false);
            acc[1][t] = __builtin_amdgcn_wmma_f32_16x16x4_f32(false, a1, false, bf[t],
                                                              (short)0, acc[1][t], false, false);
        }
    }

    // C/D layout: lanes 0-15 VGPR r -> M=r, lanes 16-31 -> M=8+r; N = l16
    #pragma unroll
    for (int mh = 0; mh < 2; ++mh) {
        const int jbase = m0 + wave * 32 + mh * 16 + 8 * half;
        #pragma unroll
        for (int t = 0; t < 4; ++t) {
            const int nb  = n0 + t * 16;
            const int cap = nb >> 5;                 // constant across 16-wide subtile
            const int kk  = (nb & 31) + l16;
            float* dst = uhat + (((size_t)b * NCAP + cap) * NIN) * DCAP + kk;
            #pragma unroll
            for (int r = 0; r < 8; ++r)
                dst[(size_t)(jbase + r) * DCAP] = acc[mh][t][r];
        }
    }
}

// ---------------- Routing: one block per batch b. 16 waves, wave i = capsule i,
// lane = capsule-dim k. Logits/coupling [16][1024] live in 64 KB LDS.
__global__ __launch_bounds__(512, 1)
void capsule_routing(const float* __restrict__ uhat, float* __restrict__ out)
{
    __shared__ float cb[NCAP * NIN];        // 64 KB, logits -> coupling in place
    const int b    = blockIdx.x;
    const int tid  = threadIdx.x;
    const int wave = tid >> 5;              // capsule index i
    const int lane = tid & 31;              // k

    const float* uhc = uhat + (((size_t)b * NCAP + wave) * NIN) * DCAP + lane;

    for (int i = tid; i < NCAP * NIN; i += 512) cb[i] = 0.0f;
    __syncthreads();

    float o = 0.0f;
    for (int it = 0; it < 4; ++it) {
        // c = softmax over capsule axis (16 values) for each j; 2 j's per thread
        for (int j = tid; j < NIN; j += 512) {
            float m = cb[j];
            #pragma unroll
            for (int i = 1; i < NCAP; ++i) m = fmaxf(m, cb[i * NIN + j]);
            float e[NCAP]; float ssum = 0.0f;
            #pragma unroll
            for (int i = 0; i < NCAP; ++i) { e[i] = __expf(cb[i * NIN + j] - m); ssum += e[i]; }
            const float inv = __frcp_rn(ssum);
            #pragma unroll
            for (int i = 0; i < NCAP; ++i) cb[i * NIN + j] = e[i] * inv;
        }
        __syncthreads();

        // s[k] = sum_j c[i][j] * u_hat[b][i][j][k]  (coalesced 128B loads, LDS broadcast)
        float s = 0.0f;
        #pragma unroll 8
        for (int j = 0; j < NIN; ++j)
            s = fmaf(cb[wave * NIN + j], uhc[(size_t)j * DCAP], s);

        // squash (unit-norm variant): wave-wide reduce of s^2 over k
        float n2 = s * s;
        #pragma unroll
        for (int m = 16; m >= 1; m >>= 1) n2 += __shfl_xor(n2, m, 32);
        o = s * rsqrtf(n2 + 1e-7f);

        if (it < 3) {
            // b[i][j] = sum_k o[k] * u_hat[b][i][j][k]
            #pragma unroll 2
            for (int j = 0; j < NIN; ++j) {
                float p = o * uhc[(size_t)j * DCAP];
                #pragma unroll
                for (int m = 16; m >= 1; m >>= 1) p += __shfl_xor(p, m, 32);
                if (lane == 0) cb[wave * NIN + j] = p;
            }
            __syncthreads();
        }
    }

    out[(size_t)b * NOUT + wave * DCAP + lane] = o;
}

// ---------------- Host-side launcher (graph-capture safe: only kernel launches)
extern "C" void kernel_launch(void* const* d_in, const int* in_sizes, int n_in,
                              void* d_out, int out_size, void* d_ws, size_t ws_size,
                              hipStream_t stream) {
    const float* x = (const float*)d_in[0];     // [128,1024,256] fp32
    const float* W = (const float*)d_in[1];     // [1,256,512] fp32
    float* out  = (float*)d_out;                // [128,16,32] fp32
    float* uhat = (float*)d_ws;                 // needs 128*16*1024*32*4 = 256 MB scratch

    dim3 ggrid(NOUT / BN, NIN / BM, BATCH);     // (8, 4, 128)
    capsule_gemm<<<ggrid, dim3(256), 0, stream>>>(x, W, uhat);
    capsule_routing<<<dim3(BATCH), dim3(512), 0, stream>>>(uhat, out);
}